// SurfaceFieldSimplification_1211180777623
// MI455X (gfx1250) — compile-verified
//
#include <hip/hip_runtime.h>
#include <cstdint>

// SurfaceFieldSimplification: out[b,p,:] = sum_k w[p,k]*verts[b,face[k],:]
//                                        + normalize(cross(v1-v0, v2-v0)) * proj_dist[p]
//
// Memory-bound gather kernel (~0.6 FLOP/byte). One thread per point p, loop
// over batches b: per-point metadata is loaded once (non-temporal — streamed
// exactly once, keep it out of L2) and reused for all B batches, while the
// vertex gathers for batch b all land in the same 4.8 MB verts slice that we
// deliberately keep L2-resident (192 MB L2 vs 154 MB verts). Output is
// write-once -> non-temporal stores so it doesn't evict verts from L2.
// gfx1250 paths: global_prefetch_b8 (next-batch gathered vertices),
// TH_NT vmem ops, wave32 codegen, merged global_load_b96 gathers.

__global__ __launch_bounds__(256) void surface_field_kernel(
    const float* __restrict__ verts,   // (B, V, 3)
    const float* __restrict__ w2d,     // (P, 3)
    const float* __restrict__ pdist,   // (P,)
    const int*   __restrict__ faces,   // (F, 3)
    const int*   __restrict__ nfid,    // (P,)
    float*       __restrict__ out,     // (B, P, 3)
    int B, long long V3, int P)
{
    int p = blockIdx.x * blockDim.x + threadIdx.x;
    if (p >= P) return;

    // Per-point metadata: read exactly once per kernel -> non-temporal loads
    // (TH=NT) so these streams don't displace verts in L2.
    long long f  = (long long)__builtin_nontemporal_load(nfid + p);
    long long i0 = (long long)__builtin_nontemporal_load(faces + 3 * f + 0) * 3;
    long long i1 = (long long)__builtin_nontemporal_load(faces + 3 * f + 1) * 3;
    long long i2 = (long long)__builtin_nontemporal_load(faces + 3 * f + 2) * 3;
    float w0 = __builtin_nontemporal_load(w2d + 3 * (long long)p + 0);
    float w1 = __builtin_nontemporal_load(w2d + 3 * (long long)p + 1);
    float w2 = __builtin_nontemporal_load(w2d + 3 * (long long)p + 2);
    float d  = __builtin_nontemporal_load(pdist + p);

    const float* vb = verts;
    float*       ob = out + 3LL * (long long)p;

    for (int b = 0; b < B; ++b) {
        // Prefetch next batch's gathered vertices (emits global_prefetch_b8).
        if (b + 1 < B) {
            const float* vn = vb + V3;
            __builtin_prefetch(vn + i0, 0, 1);
            __builtin_prefetch(vn + i1, 0, 1);
            __builtin_prefetch(vn + i2, 0, 1);
        }

        // Gather the 3 face vertices (9 floats, merged b96 loads) — these are
        // the re-used data; default temporal hint keeps them cached in L2.
        float ax = vb[i0 + 0], ay = vb[i0 + 1], az = vb[i0 + 2];
        float bx = vb[i1 + 0], by = vb[i1 + 1], bz = vb[i1 + 2];
        float cx = vb[i2 + 0], cy = vb[i2 + 1], cz = vb[i2 + 2];

        // Barycentric projection point.
        float px = fmaf(w0, ax, fmaf(w1, bx, w2 * cx));
        float py = fmaf(w0, ay, fmaf(w1, by, w2 * cy));
        float pz = fmaf(w0, az, fmaf(w1, bz, w2 * cz));

        // Face normal = cross(v1 - v0, v2 - v0), normalized.
        float e1x = bx - ax, e1y = by - ay, e1z = bz - az;
        float e2x = cx - ax, e2y = cy - ay, e2z = cz - az;
        float nx = e1y * e2z - e1z * e2y;
        float ny = e1z * e2x - e1x * e2z;
        float nz = e1x * e2y - e1y * e2x;
        float nn = nx * nx + ny * ny + nz * nz;
        float s  = d / sqrtf(nn);   // proj_dist * (1 / ||n||)

        // Output is write-once, never re-read -> non-temporal stores (TH=NT)
        // so the 38 MB result stream doesn't evict verts from L2.
        __builtin_nontemporal_store(fmaf(nx, s, px), ob + 0);
        __builtin_nontemporal_store(fmaf(ny, s, py), ob + 1);
        __builtin_nontemporal_store(fmaf(nz, s, pz), ob + 2);

        vb += V3;                 // next batch's verts slice
        ob += 3LL * (long long)P; // next batch's output slice
    }
}

extern "C" void kernel_launch(void* const* d_in, const int* in_sizes, int n_in,
                              void* d_out, int out_size, void* d_ws, size_t ws_size,
                              hipStream_t stream) {
    // Input order (setup_inputs): verts, weights_2d, proj_dist, faces, nearest_face_id
    const float* verts = (const float*)d_in[0];
    const float* w2d   = (const float*)d_in[1];
    const float* pdist = (const float*)d_in[2];
    const int*   faces = (const int*)d_in[3];
    const int*   nfid  = (const int*)d_in[4];
    float*       out   = (float*)d_out;

    const int P = in_sizes[2];                       // proj_dist has P elements
    const int B = out_size / (3 * P);                // out is (B, P, 3)
    const long long V3 = (long long)in_sizes[0] / B; // verts is (B, V, 3) -> V*3

    const int threads = 256;                          // 8 wave32 waves per block
    const int blocks  = (P + threads - 1) / threads;
    surface_field_kernel<<<blocks, threads, 0, stream>>>(
        verts, w2d, pdist, faces, nfid, out, B, V3, P);
}